// MoEGate_10557029614152
// MI455X (gfx1250) — compile-verified
//
#include <hip/hip_runtime.h>
#include <hip/hip_bf16.h>

typedef __attribute__((ext_vector_type(2))) float v2f;
typedef __attribute__((ext_vector_type(8))) float v8f;

#define NUM_E    8
#define SEQ_LEN  8192
#define BSZ      128
#define NDEV     4
#define HDIM     12
#define TOK_TOTAL (BSZ * SEQ_LEN)          // 1,048,576 tokens
#define ALPHA2   1.0e-3f
#define ALPHA3   1.0e-3f

// ---------------------------------------------------------------------------
// Zero the per-batch accumulator workspace (graph-safe re-init every call)
// ---------------------------------------------------------------------------
__global__ __launch_bounds__(256) void moe_zero_ws(float* __restrict__ p, int n) {
  int i = blockIdx.x * blockDim.x + threadIdx.x;
  if (i < n) p[i] = 0.0f;
}

// ---------------------------------------------------------------------------
// Main kernel: WMMA gate GEMM + softmax + top2 + per-batch stat accumulation.
// Block = 256 threads = 8 waves; each wave handles 32 tokens (two 16-token
// WMMA tiles). 32 blocks per batch (8192 tokens), grid = 4096 blocks.
// ---------------------------------------------------------------------------
__global__ __launch_bounds__(256) void moe_gate_main(
    const float* __restrict__ x,        // [T, 12]
    const float* __restrict__ w,        // [8, 12]
    float* __restrict__ out_idx,        // [T, 2]  (indices as float)
    float* __restrict__ out_wt,         // [T, 2]
    float* __restrict__ wsScore,        // [B, 8]  sum of softmax scores
    float* __restrict__ wsCnt,          // [B, 8]  top-2 hit counts
    float* __restrict__ wsDev)          // [B, 4]  device-group member counts
{
  const int lane    = threadIdx.x & 31;
  const int wv      = threadIdx.x >> 5;
  const int tokBase = blockIdx.x * 256 + wv * 32;
  const int batch   = blockIdx.x >> 5;           // 8192 / 256 = 32 blocks/batch

  const int colSel = lane & 15;                  // N (token within tile)
  const int hi     = lane >> 4;                  // lane half
  const int kb     = hi * 2;                     // K sub-offset within chunk
  const int expert = lane & 7;                   // A rows: experts duplicated

  // ---- A fragments: gate weight [8,12], rows 8..15 duplicate experts 0..7.
  // 16x4 f32 A layout: VGPR0 = K={0|2}, VGPR1 = K={1|3} by lane half.
  const float* wr = w + expert * HDIM + kb;
  v2f a0, a1, a2;
  a0.x = wr[0]; a0.y = wr[1];                    // K chunk 0 (k = 0..3)
  a1.x = wr[4]; a1.y = wr[5];                    // K chunk 1 (k = 4..7)
  a2.x = wr[8]; a2.y = wr[9];                    // K chunk 2 (k = 8..11)

  // ---- B fragments: B = x^T (K x N); two 16-token tiles per wave.
  const float* xt0 = x + (size_t)(tokBase +      colSel) * HDIM + kb;
  const float* xt1 = x + (size_t)(tokBase + 16 + colSel) * HDIM + kb;
  v2f b00, b01, b02, b10, b11, b12;
  b00.x = xt0[0]; b00.y = xt0[1];
  b01.x = xt0[4]; b01.y = xt0[5];
  b02.x = xt0[8]; b02.y = xt0[9];
  b10.x = xt1[0]; b10.y = xt1[1];
  b11.x = xt1[4]; b11.y = xt1[5];
  b12.x = xt1[8]; b12.y = xt1[9];

  // ---- logits^T = A x B, K=12 via 3 chained 16x16x4 f32 WMMAs per tile.
  v8f c0 = {};
  c0 = __builtin_amdgcn_wmma_f32_16x16x4_f32(false, a0, false, b00, (short)0, c0, false, false);
  c0 = __builtin_amdgcn_wmma_f32_16x16x4_f32(false, a1, false, b01, (short)0, c0, false, false);
  c0 = __builtin_amdgcn_wmma_f32_16x16x4_f32(false, a2, false, b02, (short)0, c0, false, false);
  v8f c1 = {};
  c1 = __builtin_amdgcn_wmma_f32_16x16x4_f32(false, a0, false, b10, (short)0, c1, false, false);
  c1 = __builtin_amdgcn_wmma_f32_16x16x4_f32(false, a1, false, b11, (short)0, c1, false, false);
  c1 = __builtin_amdgcn_wmma_f32_16x16x4_f32(false, a2, false, b12, (short)0, c1, false, false);

  // With duplicated expert rows, D VGPR i holds logit[expert i][token col] in
  // BOTH lane halves. Lane L owns token tokBase + L: lanes 0-15 take tile 0,
  // lanes 16-31 take tile 1 (their col = L&15 -> token tokBase+16+(L&15)).
  float lg[NUM_E];
#pragma unroll
  for (int i = 0; i < NUM_E; ++i) lg[i] = hi ? c1[i] : c0[i];

  // ---- softmax over 8 experts (lane-local)
  float m = lg[0];
#pragma unroll
  for (int i = 1; i < NUM_E; ++i) m = fmaxf(m, lg[i]);
  float ex[NUM_E], sum = 0.0f;
#pragma unroll
  for (int i = 0; i < NUM_E; ++i) { ex[i] = __expf(lg[i] - m); sum += ex[i]; }
  const float inv = 1.0f / sum;
  float sc[NUM_E];
#pragma unroll
  for (int i = 0; i < NUM_E; ++i) sc[i] = ex[i] * inv;

  // ---- top-2 (first-occurrence tie-breaking, matching lax.top_k)
  int i0 = 0; float s0 = sc[0];
#pragma unroll
  for (int i = 1; i < NUM_E; ++i) { if (sc[i] > s0) { s0 = sc[i]; i0 = i; } }
  int i1 = -1; float s1 = -1.0f;
#pragma unroll
  for (int i = 0; i < NUM_E; ++i) { if (i != i0 && sc[i] > s1) { s1 = sc[i]; i1 = i; } }

  // ---- write outputs (8B/lane contiguous -> fully coalesced b64 stores)
  const int t = tokBase + lane;
  out_idx[2 * t + 0] = (float)i0;
  out_idx[2 * t + 1] = (float)i1;
  out_wt [2 * t + 0] = s0;
  out_wt [2 * t + 1] = s1;

  // ---- per-wave reductions
  // score sums (for Pi): butterfly across the wave
  float ssum[NUM_E];
#pragma unroll
  for (int e = 0; e < NUM_E; ++e) {
    float v = sc[e];
    for (int o = 16; o > 0; o >>= 1) v += __shfl_xor(v, o, 32);
    ssum[e] = v;
  }
  // expert hit counts and device-group membership via ballots (free cross-lane)
  float cntE[NUM_E];
#pragma unroll
  for (int e = 0; e < NUM_E; ++e)
    cntE[e] = (float)(__popcll(__ballot(i0 == e)) + __popcll(__ballot(i1 == e)));
  const int d0 = i0 >> 1, d1 = i1 >> 1;          // step = E/D = 2
  float cntD[NDEV];
#pragma unroll
  for (int d = 0; d < NDEV; ++d)
    cntD[d] = (float)__popcll(__ballot(d0 == d || d1 == d));

  // ---- block-level combine in LDS, then one global atomic per accumulator
  __shared__ float sS[NUM_E];
  __shared__ float sC[NUM_E];
  __shared__ float sD[NDEV];
  if (threadIdx.x < NUM_E) { sS[threadIdx.x] = 0.0f; sC[threadIdx.x] = 0.0f; }
  if (threadIdx.x < NDEV)  { sD[threadIdx.x] = 0.0f; }
  __syncthreads();
  if (lane == 0) {
#pragma unroll
    for (int e = 0; e < NUM_E; ++e) {
      atomicAdd(&sS[e], ssum[e]);
      atomicAdd(&sC[e], cntE[e]);
    }
#pragma unroll
    for (int d = 0; d < NDEV; ++d) atomicAdd(&sD[d], cntD[d]);
  }
  __syncthreads();
  if (threadIdx.x < NUM_E) {
    atomicAdd(&wsScore[batch * NUM_E + threadIdx.x], sS[threadIdx.x]);
    atomicAdd(&wsCnt  [batch * NUM_E + threadIdx.x], sC[threadIdx.x]);
  }
  if (threadIdx.x < NDEV) {
    atomicAdd(&wsDev[batch * NDEV + threadIdx.x], sD[threadIdx.x]);
  }
}

// ---------------------------------------------------------------------------
// Loss kernel: one block, one thread per batch; fold 128 batches -> 2 scalars
// ---------------------------------------------------------------------------
__global__ __launch_bounds__(128) void moe_gate_loss(
    const float* __restrict__ wsScore, const float* __restrict__ wsCnt,
    const float* __restrict__ wsDev, float* __restrict__ out /* [2] */)
{
  const int b    = threadIdx.x;              // 0..127
  const int lane = threadIdx.x & 31;
  const int wv   = threadIdx.x >> 5;

  const float ceScale = (float)NUM_E / (float)(SEQ_LEN * 2);   // E/(S*k)
  const float piScale = 1.0f / (float)SEQ_LEN;
  const float nwScale = (float)NDEV / (float)SEQ_LEN;          // D/(S*M)

  float dev = 0.0f, comm = 0.0f;
#pragma unroll
  for (int d = 0; d < NDEV; ++d) {
    float ce0 = wsCnt[b * NUM_E + 2 * d + 0] * ceScale;
    float ce1 = wsCnt[b * NUM_E + 2 * d + 1] * ceScale;
    float fg  = 0.5f * (ce0 + ce1);                            // group mean
    float Pg  = (wsScore[b * NUM_E + 2 * d + 0] +
                 wsScore[b * NUM_E + 2 * d + 1]) * piScale;    // group sum
    float nw  = wsDev[b * NDEV + d] * nwScale;
    dev  += fg * Pg;
    comm += nw * Pg;
  }
  // reduce 128 -> 1
  for (int o = 16; o > 0; o >>= 1) {
    dev  += __shfl_xor(dev,  o, 32);
    comm += __shfl_xor(comm, o, 32);
  }
  __shared__ float rD[4], rC[4];
  if (lane == 0) { rD[wv] = dev; rC[wv] = comm; }
  __syncthreads();
  if (threadIdx.x == 0) {
    float D = 0.0f, C = 0.0f;
#pragma unroll
    for (int i = 0; i < 4; ++i) { D += rD[i]; C += rC[i]; }
    out[0] = D * (ALPHA2 / (float)BSZ);
    out[1] = C * (ALPHA3 / (float)BSZ);
  }
}

// ---------------------------------------------------------------------------
extern "C" void kernel_launch(void* const* d_in, const int* in_sizes, int n_in,
                              void* d_out, int out_size, void* d_ws, size_t ws_size,
                              hipStream_t stream) {
  const float* x = (const float*)d_in[0];   // [128, 8192, 12]
  const float* w = (const float*)d_in[1];   // [8, 12]
  float* out = (float*)d_out;

  const size_t T = (size_t)TOK_TOTAL;
  float* out_idx = out;                     // [T,2]
  float* out_wt  = out + 2 * T;             // [T,2]
  float* out_sc  = out + 4 * T;             // [2] losses

  float* ws      = (float*)d_ws;
  float* wsScore = ws;                      // [128,8]
  float* wsCnt   = ws + BSZ * NUM_E;        // [128,8]
  float* wsDev   = ws + 2 * BSZ * NUM_E;    // [128,4]
  const int nws  = BSZ * (NUM_E + NUM_E + NDEV);   // 2560 floats

  moe_zero_ws<<<(nws + 255) / 256, 256, 0, stream>>>(ws, nws);
  moe_gate_main<<<TOK_TOTAL / 256, 256, 0, stream>>>(x, w, out_idx, out_wt,
                                                     wsScore, wsCnt, wsDev);
  moe_gate_loss<<<1, 128, 0, stream>>>(wsScore, wsCnt, wsDev, out_sc);
}